// MDTA_6107443495403
// MI455X (gfx1250) — compile-verified
//
#include <hip/hip_runtime.h>

// MDTA (Restormer transposed attention) for gfx1250 / MI455X.
// All matrix stages use V_WMMA_F32_16X16X4_F32 (native fp32 WMMA path).
//
// Shapes: B=8, C=192, H=W=128 (HW=16384), heads=8, head_dim=24.

typedef float v2f __attribute__((ext_vector_type(2)));
typedef float v8f __attribute__((ext_vector_type(8)));

#define HW_   16384
#define Kc    192     // channels (inner dim of the 1x1 convs)
#define Bn    8
#define NH    8
#define HD    24      // head dim

static __device__ __forceinline__ v8f wmma_f32(v2f a, v2f b, v8f c) {
  // 8 args: (neg_a, A, neg_b, B, c_mod, C, reuse_a, reuse_b)
  return __builtin_amdgcn_wmma_f32_16x16x4_f32(false, a, false, b, (short)0, c,
                                               false, false);
}
static __device__ __forceinline__ v2f v2zero() { v2f z; z.x = 0.f; z.y = 0.f; return z; }
static __device__ __forceinline__ v8f v8zero() {
  v8f z = {0.f,0.f,0.f,0.f,0.f,0.f,0.f,0.f}; return z;
}

// ---------------------------------------------------------------------------
// Kernel 0: zero S + norms accumulators (they are atomically accumulated)
// ---------------------------------------------------------------------------
__global__ void zero_kernel(float* __restrict__ p, int n) {
  int i = blockIdx.x * blockDim.x + threadIdx.x;
  if (i < n) p[i] = 0.f;
}

// ---------------------------------------------------------------------------
// Pointwise (1x1 conv) GEMM: Y[b][o][p] = sum_c W[o][c] * X[b][c][p]
// K = 192 fixed. One wave computes a 16(M) x 64(N) tile, K-steps of 4.
// Used for both the qkv projection (O=576) and the output projection (O=192).
// ---------------------------------------------------------------------------
__global__ __launch_bounds__(256)
void pw_gemm_kernel(const float* __restrict__ W, const float* __restrict__ X,
                    float* __restrict__ Y, int O) {
  const int lane = threadIdx.x & 31;
  const int wid  = threadIdx.x >> 5;
  const int jw   = blockIdx.x * 8 + wid;

  const int ntiles = HW_ / 64;            // 256
  const int mtiles = O / 16;
  const int jobsPerBatch = mtiles * ntiles;
  const int b  = jw / jobsPerBatch;
  const int r  = jw - b * jobsPerBatch;
  const int m0 = (r / ntiles) * 16;
  const int n0 = (r % ntiles) * 64;

  const int row = lane & 15;              // A: M row / B,D: N col
  const int kh  = (lane >> 4) * 2;        // K sub-offset per half-wave

  v8f acc0 = v8zero(), acc1 = v8zero(), acc2 = v8zero(), acc3 = v8zero();

  const float* wrowp = W + (size_t)(m0 + row) * Kc + kh;
  const float* xbase = X + ((size_t)b * Kc + kh) * HW_ + n0 + row;

  for (int k = 0; k < Kc; k += 4) {
    v2f a = *(const v2f*)(wrowp + k);                 // W[m][k+kh], W[m][k+kh+1]
    const float* xp = xbase + (size_t)k * HW_;
    v2f b0, b1, b2, b3;
    b0.x = xp[0];   b0.y = xp[HW_];                   // X rows k+kh, k+kh+1
    b1.x = xp[16];  b1.y = xp[HW_ + 16];
    b2.x = xp[32];  b2.y = xp[HW_ + 32];
    b3.x = xp[48];  b3.y = xp[HW_ + 48];
    acc0 = wmma_f32(a, b0, acc0);
    acc1 = wmma_f32(a, b1, acc1);
    acc2 = wmma_f32(a, b2, acc2);
    acc3 = wmma_f32(a, b3, acc3);
  }

  const int mhi = (lane >> 4) * 8;
  float* ybase = Y + ((size_t)b * O + m0 + mhi) * HW_ + n0 + row;
#pragma unroll
  for (int rr = 0; rr < 8; ++rr) {
    float* yp = ybase + (size_t)rr * HW_;
    yp[0]  = acc0[rr];
    yp[16] = acc1[rr];
    yp[32] = acc2[rr];
    yp[48] = acc3[rr];
  }
}

// ---------------------------------------------------------------------------
// Depthwise 3x3 conv (padding=1) + fused sum-of-squares for q,k channel norms.
// grid.x = B*576 (one channel plane), grid.y = 64 (256-pixel chunks).
// ---------------------------------------------------------------------------
__global__ __launch_bounds__(256)
void dwconv_kernel(const float* __restrict__ in, const float* __restrict__ wdw,
                   float* __restrict__ out, float* __restrict__ norms) {
  const int bc = blockIdx.x;
  const int b  = bc / 576;
  const int ch = bc - b * 576;
  const int p  = blockIdx.y * 256 + threadIdx.x;
  const int y  = p >> 7;
  const int x  = p & 127;
  const float* plane = in + (size_t)bc * HW_;

  float w[9];
#pragma unroll
  for (int i = 0; i < 9; ++i) w[i] = wdw[ch * 9 + i];

  float acc = 0.f;
#pragma unroll
  for (int dy = -1; dy <= 1; ++dy) {
    int yy = y + dy;
    if (yy < 0 || yy > 127) continue;
#pragma unroll
    for (int dx = -1; dx <= 1; ++dx) {
      int xx = x + dx;
      if (xx < 0 || xx > 127) continue;
      acc += w[(dy + 1) * 3 + (dx + 1)] * plane[yy * 128 + xx];
    }
  }
  out[(size_t)bc * HW_ + p] = acc;

  // fused ||q_c||^2 / ||k_c||^2 accumulation (channels 0..383)
  __shared__ float red[256];
  red[threadIdx.x] = (ch < 384) ? acc * acc : 0.f;
  __syncthreads();
  for (int s = 128; s > 0; s >>= 1) {
    if (threadIdx.x < s) red[threadIdx.x] += red[threadIdx.x + s];
    __syncthreads();
  }
  if (threadIdx.x == 0 && ch < 384) atomicAdd(&norms[b * 384 + ch], red[0]);
}

// ---------------------------------------------------------------------------
// Gram matrix: S[bh][c][d] = sum_n q[c][n] * k[d][n]   (c,d < 24, n < 16384)
// 24 padded to 32 -> 2x2 WMMA tiles; K (=n) split over 16 waves per (b,h),
// partials accumulated with atomicAdd.
// ---------------------------------------------------------------------------
__global__ __launch_bounds__(256)
void gram_kernel(const float* __restrict__ qkv, float* __restrict__ S) {
  const int lane = threadIdx.x & 31;
  const int wid  = threadIdx.x >> 5;
  const int jw   = blockIdx.x * 8 + wid;   // 64 bh * 16 slices = 1024 jobs
  const int bh = jw >> 4;
  const int sl = jw & 15;
  const int b = bh >> 3, h = bh & 7;
  const int nbase = sl * 1024;

  const float* qb = qkv + ((size_t)b * 576 + h * HD) * HW_;         // q planes
  const float* kb = qkv + ((size_t)b * 576 + 192 + h * HD) * HW_;   // k planes

  const int row = lane & 15;
  const int kh  = (lane >> 4) * 2;
  const bool hi_valid = (row < 8);   // rows 16..23 of the padded 32

  v8f s00 = v8zero(), s01 = v8zero(), s10 = v8zero(), s11 = v8zero();

  for (int n = nbase; n < nbase + 1024; n += 4) {
    v2f a0 = *(const v2f*)(qb + (size_t)row * HW_ + n + kh);
    v2f b0 = *(const v2f*)(kb + (size_t)row * HW_ + n + kh);
    v2f a1 = hi_valid ? *(const v2f*)(qb + (size_t)(16 + row) * HW_ + n + kh)
                      : v2zero();
    v2f b1 = hi_valid ? *(const v2f*)(kb + (size_t)(16 + row) * HW_ + n + kh)
                      : v2zero();
    s00 = wmma_f32(a0, b0, s00);
    s01 = wmma_f32(a0, b1, s01);
    s10 = wmma_f32(a1, b0, s10);
    s11 = wmma_f32(a1, b1, s11);
  }

  const int mhi = (lane >> 4) * 8;
  float* Sb = S + (size_t)bh * (HD * HD);
#pragma unroll
  for (int rr = 0; rr < 8; ++rr) {
    const int c = rr + mhi;   // 0..15
    const int d = row;        // 0..15
    atomicAdd(&Sb[c * HD + d], s00[rr]);
    if (16 + d < HD) atomicAdd(&Sb[c * HD + 16 + d], s01[rr]);
    if (16 + c < HD) {
      atomicAdd(&Sb[(16 + c) * HD + d], s10[rr]);
      if (16 + d < HD) atomicAdd(&Sb[(16 + c) * HD + 16 + d], s11[rr]);
    }
  }
}

// ---------------------------------------------------------------------------
// Softmax over the 24x24 attention logits with L2-norm + temperature scaling.
// One block (1 wave) per (b,h); thread c < 24 handles one row.
// ---------------------------------------------------------------------------
__global__ void softmax_kernel(const float* __restrict__ S,
                               const float* __restrict__ norms,
                               const float* __restrict__ temp,
                               float* __restrict__ attn) {
  const int bh = blockIdx.x;
  const int b = bh >> 3, h = bh & 7;
  const int c = threadIdx.x;
  if (c >= HD) return;
  const float eps = 1e-12f;
  const float qn = fmaxf(sqrtf(norms[b * 384 + h * HD + c]), eps);
  const float t  = temp[h];

  float sv[HD];
  float mx = -3.402823466e38f;
#pragma unroll
  for (int d = 0; d < HD; ++d) {
    float kn = fmaxf(sqrtf(norms[b * 384 + 192 + h * HD + d]), eps);
    float s  = S[(size_t)bh * HD * HD + c * HD + d] / (qn * kn) * t;
    sv[d] = s;
    mx = fmaxf(mx, s);
  }
  float sum = 0.f;
#pragma unroll
  for (int d = 0; d < HD; ++d) { sv[d] = __expf(sv[d] - mx); sum += sv[d]; }
  const float inv = 1.f / sum;
#pragma unroll
  for (int d = 0; d < HD; ++d)
    attn[(size_t)bh * HD * HD + c * HD + d] = sv[d] * inv;
}

// ---------------------------------------------------------------------------
// out[b][h*24+c][n] = sum_d attn[bh][c][d] * v[d][n]    (M=24 pad 32, K=24)
// One wave per 16-column strip; 2 M-tiles, 6 K-steps of 4.
// ---------------------------------------------------------------------------
__global__ __launch_bounds__(256)
void attnv_kernel(const float* __restrict__ attn, const float* __restrict__ qkv,
                  float* __restrict__ out) {
  const int lane = threadIdx.x & 31;
  const int wid  = threadIdx.x >> 5;
  const int bh = blockIdx.y;
  const int b = bh >> 3, h = bh & 7;
  const int n0 = blockIdx.x * 128 + wid * 16;

  const float* vb = qkv + ((size_t)b * 576 + 384 + h * HD) * HW_;   // v planes
  const float* Ab = attn + (size_t)bh * HD * HD;

  const int row = lane & 15;
  const int kh  = (lane >> 4) * 2;

  v8f acc0 = v8zero(), acc1 = v8zero();
  for (int k = 0; k < HD; k += 4) {
    const int kk = k + kh;               // <= 23 always
    v2f a0; a0.x = Ab[row * HD + kk]; a0.y = Ab[row * HD + kk + 1];
    v2f a1 = v2zero();
    if (row < 8) { a1.x = Ab[(16 + row) * HD + kk]; a1.y = Ab[(16 + row) * HD + kk + 1]; }
    const float* vp = vb + (size_t)kk * HW_ + n0 + row;
    v2f bb; bb.x = vp[0]; bb.y = vp[HW_];
    acc0 = wmma_f32(a0, bb, acc0);
    acc1 = wmma_f32(a1, bb, acc1);
  }

  const int mhi = (lane >> 4) * 8;
  float* ob = out + ((size_t)b * 192 + h * HD) * HW_ + n0 + row;
#pragma unroll
  for (int rr = 0; rr < 8; ++rr) {
    const int c = rr + mhi;              // 0..15 always valid for tile 0
    ob[(size_t)c * HW_] = acc0[rr];
    if (16 + c < HD) ob[(size_t)(16 + c) * HW_] = acc1[rr];
  }
}

// ---------------------------------------------------------------------------
// Host launcher
// ---------------------------------------------------------------------------
extern "C" void kernel_launch(void* const* d_in, const int* in_sizes, int n_in,
                              void* d_out, int out_size, void* d_ws, size_t ws_size,
                              hipStream_t stream) {
  const float* x      = (const float*)d_in[0];   // (8,192,128,128)
  const float* w_qkv  = (const float*)d_in[1];   // (576,192)
  const float* w_dw   = (const float*)d_in[2];   // (576,1,3,3)
  const float* w_proj = (const float*)d_in[3];   // (192,192)
  const float* temp   = (const float*)d_in[4];   // (8,1,1)
  float* out = (float*)d_out;

  float* ws = (float*)d_ws;
  const size_t QKV = (size_t)Bn * 576 * HW_;     // 75,497,472 floats
  float* qkv_pre = ws;                           // reused as attn_out later
  float* qkv_dw  = ws + QKV;
  float* Sbuf    = ws + 2 * QKV;                 // 64*24*24
  float* norms   = Sbuf + 64 * HD * HD;          // 8*384
  float* attnbuf = norms + Bn * 384;             // 64*24*24
  float* attn_out = qkv_pre;                     // alias (qkv_pre dead by then)

  const int nzero = 64 * HD * HD + Bn * 384;     // S + norms (contiguous)
  zero_kernel<<<(nzero + 255) / 256, 256, 0, stream>>>(Sbuf, nzero);

  // 1) qkv = W_qkv @ x         : jobs = 8 * (576/16) * (16384/64) = 73728
  pw_gemm_kernel<<<Bn * 36 * 256 / 8, 256, 0, stream>>>(w_qkv, x, qkv_pre, 576);

  // 2) depthwise 3x3 + q/k sum-of-squares
  dwconv_kernel<<<dim3(Bn * 576, 64), 256, 0, stream>>>(qkv_pre, w_dw, qkv_dw, norms);

  // 3) S = q @ k^T (per b,h)   : 1024 wave jobs
  gram_kernel<<<128, 256, 0, stream>>>(qkv_dw, Sbuf);

  // 4) softmax with norm + temperature scaling
  softmax_kernel<<<64, 32, 0, stream>>>(Sbuf, norms, temp, attnbuf);

  // 5) out_attn = attn @ v
  attnv_kernel<<<dim3(128, 64), 256, 0, stream>>>(attnbuf, qkv_dw, attn_out);

  // 6) final 1x1 projection    : jobs = 8 * (192/16) * (16384/64) = 24576
  pw_gemm_kernel<<<Bn * 12 * 256 / 8, 256, 0, stream>>>(w_proj, attn_out, out, 192);
}